// HiPool_3590592659529
// MI455X (gfx1250) — compile-verified
//
#include <hip/hip_runtime.h>
#include <stdint.h>

// HiPool: multi-scale max-pool (11 dyadic scales) + softmax-weighted combine.
// Memory-bound: 553 MB stream -> ~24us floor at 23.3 TB/s.
// CDNA5 path: async global->LDS triple-buffered pipeline (ASYNCcnt), wave32.

#define B_DIM    256
#define T_DIM    1024
#define C_DIM    527
#define SPLIT    2
#define CHUNK    (T_DIM / SPLIT)      // 512 timesteps per block
#define ROWS     8                    // timesteps per pipeline group
#define NGROUPS  (CHUNK / ROWS)       // 64
#define STAGES   3                    // LDS pipeline depth
#define PADC     544                  // LDS row stride in dwords (== blockDim)
#define NTHREADS 544                  // 17 wave32
#define ROWBYTES (C_DIM * 4)          // 2108

// Generic pointer -> 32-bit LDS byte offset (AS3 pointer value == LDS offset).
__device__ __forceinline__ uint32_t lds_off_u32(const void* p) {
  return (uint32_t)(uintptr_t)(__attribute__((address_space(3))) const void*)p;
}

__global__ __launch_bounds__(NTHREADS) void hipool_partial(
    const float* __restrict__ y, const float* __restrict__ w,
    float* __restrict__ contrib, float* __restrict__ chunkmax) {
  __shared__ float smem[STAGES * ROWS * PADC];

  const int tid  = threadIdx.x;
  const int half = blockIdx.x;   // which T chunk
  const int b    = blockIdx.y;   // batch
  const int c    = (tid < C_DIM) ? tid : (C_DIM - 1);   // clamp, no divergence

  const uint32_t ldsbase = lds_off_u32(smem) + (uint32_t)tid * 4u;
  const uint32_t gbase =
      ((uint32_t)(b * T_DIM + half * CHUNK) * (uint32_t)C_DIM + (uint32_t)c) * 4u;

  // Issue one group (8 rows) of async global->LDS b32 loads into its stage.
  auto issue = [&](int g) {
    const uint32_t l0 = ldsbase + (uint32_t)(g % STAGES) * (uint32_t)(ROWS * PADC * 4);
    const uint32_t g0 = gbase + (uint32_t)g * (uint32_t)(ROWS * ROWBYTES);
#pragma unroll
    for (int r = 0; r < ROWS; ++r) {
      uint32_t l  = l0 + (uint32_t)(r * PADC * 4);
      uint32_t go = g0 + (uint32_t)(r * ROWBYTES);
      asm volatile("global_load_async_to_lds_b32 %0, %1, %2 offset:0"
                   :: "v"(l), "v"(go), "s"(y)
                   : "memory");
    }
  };

  issue(0);
  issue(1);

  float accs[10];                 // accs[k] = sum of bin-maxes, bin size 2^k
#pragma unroll
  for (int k = 0; k < 10; ++k) accs[k] = 0.0f;
  float part[6];                  // carry-propagate partial maxes (bins 16..512)

  for (int g = 0; g < NGROUPS; ++g) {
    if (g + (STAGES - 1) < NGROUPS) issue(g + (STAGES - 1));
    // In-order async completion: wait for group g (2 stages stay in flight).
    if (g < NGROUPS - 2)       asm volatile("s_wait_asynccnt 16" ::: "memory");
    else if (g == NGROUPS - 2) asm volatile("s_wait_asynccnt 8"  ::: "memory");
    else                       asm volatile("s_wait_asynccnt 0"  ::: "memory");

    const float* s = &smem[(g % STAGES) * (ROWS * PADC) + tid];
    float x0 = s[0 * PADC], x1 = s[1 * PADC], x2 = s[2 * PADC], x3 = s[3 * PADC];
    float x4 = s[4 * PADC], x5 = s[5 * PADC], x6 = s[6 * PADC], x7 = s[7 * PADC];

    accs[0] += ((x0 + x1) + (x2 + x3)) + ((x4 + x5) + (x6 + x7));
    float m01 = fmaxf(x0, x1), m23 = fmaxf(x2, x3);
    float m45 = fmaxf(x4, x5), m67 = fmaxf(x6, x7);
    accs[1] += (m01 + m23) + (m45 + m67);
    float m03 = fmaxf(m01, m23), m47 = fmaxf(m45, m67);
    accs[2] += m03 + m47;
    float m8 = fmaxf(m03, m47);
    accs[3] += m8;

    // Binary-counter merge across groups: bin size 2^(4+k) completes when
    // the low k+1 bits of g are all ones.
    float m = m8;
#pragma unroll
    for (int k = 0; k < 6; ++k) {
      if (((g >> k) & 1) == 0) { part[k] = m; break; }
      m = fmaxf(part[k], m);
      accs[4 + k] += m;
    }
  }
  // accs[9] == max over this 512-chunk (single bin).

  // Per-channel softmax over the 11 scale weights (N_STAR[i] = 2^i).
  float wv[11];
#pragma unroll
  for (int i = 0; i < 11; ++i) wv[i] = w[c * 11 + i];
  float mx = wv[0];
#pragma unroll
  for (int i = 1; i < 11; ++i) mx = fmaxf(mx, wv[i]);
  float den = 0.0f;
#pragma unroll
  for (int i = 0; i < 11; ++i) { wv[i] = expf(wv[i] - mx); den += wv[i]; }
  const float inv = 1.0f / den;

  // Weighted contribution of all additive scales (bin sizes 2^0..2^9,
  // i.e. r = 1024..2). pooled(r=2^i) = total_bin_max_sum / 2^i, i = 10-k.
  float o = 0.0f;
#pragma unroll
  for (int k = 0; k < 10; ++k)
    o += (wv[10 - k] * inv) * accs[k] * (1.0f / (float)(1 << (10 - k)));

  if (tid < C_DIM) {
    const int idx = (b * SPLIT + half) * C_DIM + tid;
    contrib[idx]  = o;
    chunkmax[idx] = accs[9];
  }
}

// Combine chunks: additive parts sum; r=1 scale needs the global max.
__global__ void hipool_finish(const float* __restrict__ w,
                              const float* __restrict__ contrib,
                              const float* __restrict__ chunkmax,
                              float* __restrict__ out) {
  int idx = blockIdx.x * blockDim.x + threadIdx.x;
  if (idx >= B_DIM * C_DIM) return;
  int b = idx / C_DIM;
  int c = idx - b * C_DIM;

  float wv[11];
#pragma unroll
  for (int i = 0; i < 11; ++i) wv[i] = w[c * 11 + i];
  float mx = wv[0];
#pragma unroll
  for (int i = 1; i < 11; ++i) mx = fmaxf(mx, wv[i]);
  float den = 0.0f;
#pragma unroll
  for (int i = 0; i < 11; ++i) { wv[i] = expf(wv[i] - mx); den += wv[i]; }
  const float alpha0 = wv[0] / den;   // weight for r=1 (full-sequence max)

  const int i0 = (b * SPLIT + 0) * C_DIM + c;
  const int i1 = (b * SPLIT + 1) * C_DIM + c;
  const float gm = fmaxf(chunkmax[i0], chunkmax[i1]);
  out[idx] = contrib[i0] + contrib[i1] + alpha0 * gm;
}

extern "C" void kernel_launch(void* const* d_in, const int* in_sizes, int n_in,
                              void* d_out, int out_size, void* d_ws, size_t ws_size,
                              hipStream_t stream) {
  (void)in_sizes; (void)n_in; (void)out_size; (void)ws_size;
  const float* y = (const float*)d_in[0];   // [256,1024,527] f32
  const float* w = (const float*)d_in[1];   // [527,11] f32
  float* out      = (float*)d_out;          // [256,527] f32
  float* contrib  = (float*)d_ws;                           // [256*2*527]
  float* chunkmax = contrib + (B_DIM * SPLIT * C_DIM);      // [256*2*527]

  hipool_partial<<<dim3(SPLIT, B_DIM), NTHREADS, 0, stream>>>(y, w, contrib, chunkmax);

  const int n = B_DIM * C_DIM;
  hipool_finish<<<(n + 255) / 256, 256, 0, stream>>>(w, contrib, chunkmax, out);
}